// EdgePrompt_20392504721412
// MI455X (gfx1250) — compile-verified
//
#include <hip/hip_runtime.h>

typedef float v2f __attribute__((ext_vector_type(2)));
typedef float v8f __attribute__((ext_vector_type(8)));

#define IN_DIM 128

// ---------------------------------------------------------------------------
// Kernel 1: node projection, fp32 WMMA (V_WMMA_F32_16X16X4_F32).
// Each wave computes one 16-node tile:
//   D(16x16) += A(16x4) * B(4x16) over k = 0..124 step 4
// A layout (32-bit, 16x4): VGPR0 lanes0-15 = (M=lane, K=k),   lanes16-31 = (M=lane-16, K=k+2)
//                          VGPR1 lanes0-15 = (M=lane, K=k+1), lanes16-31 = (M=lane-16, K=k+3)
// B layout (32-bit, 4x16): VGPR0 lanes0-15 = (K=k,   N=lane), lanes16-31 = (K=k+2, N=lane-16)
//                          VGPR1 lanes0-15 = (K=k+1, N=lane), lanes16-31 = (K=k+3, N=lane-16)
// Only B columns N=0 (W[:128]) and N=1 (W[128:]) are non-zero, so
// D column 0 = proj_src, D column 1 = proj_dst for the tile's 16 nodes.
// B fragments: every lane loads unconditionally from W row (nl&1) — always in
// bounds — and lanes with nl>=2 zero the value with a branch-free select
// (v_cndmask), keeping the inner loop free of exec-mask branches.
// ---------------------------------------------------------------------------
__global__ void __launch_bounds__(256)
node_proj_wmma(const float* __restrict__ x,
               const float* __restrict__ W,
               float* __restrict__ proj_src,
               float* __restrict__ proj_dst,
               int n_nodes, int n_tiles) {
  const int lane = threadIdx.x & 31;
  const int wave = threadIdx.x >> 5;
  const int tile = blockIdx.x * (blockDim.x >> 5) + wave;
  if (tile >= n_tiles) return;   // wave-uniform: EXEC stays all-ones for WMMA

  const int half = lane >> 4;          // selects K pair within the 4-chunk
  const int nl   = lane & 15;          // A: row-in-tile, B/D: column
  int row = tile * 16 + nl;            // node index this lane loads for A
  if (row >= n_nodes) row = n_nodes - 1;  // clamp (stores are guarded below)

  // Fixed per-lane base pointers; loop offsets become immediate IOFFSETs.
  const float* __restrict__ xrow = x + (size_t)row * IN_DIM + half * 2;
  const float* __restrict__ wrow = W + (nl & 1) * IN_DIM + half * 2;
  const bool active = (nl < 2);

  v8f acc = {};
  #pragma unroll 8
  for (int k = 0; k < IN_DIM; k += 4) {
    v2f a  = *(const v2f*)(xrow + k);   // global_load_b64, imm offset
    v2f w2 = *(const v2f*)(wrow + k);   // global_load_b64, imm offset (WGP$ hit)
    v2f b;
    b.x = active ? w2.x : 0.0f;         // v_cndmask, no branch
    b.y = active ? w2.y : 0.0f;
    // 8 args: (neg_a, A, neg_b, B, c_mod, C, reuse_a, reuse_b)
    acc = __builtin_amdgcn_wmma_f32_16x16x4_f32(false, a, false, b,
                                                (short)0, acc, false, false);
  }

  // D layout: VGPR r, lanes0-15 -> (M=r, N=lane); lanes16-31 -> (M=8+r, N=lane-16)
  // Lanes with N-column 0/1 write the two projections (8 rows each per half).
  if (nl < 2) {
    float* __restrict__ dstp = (nl == 0) ? proj_src : proj_dst;
    const int mbase = tile * 16 + half * 8;
    #pragma unroll
    for (int r = 0; r < 8; ++r) {
      const int m = mbase + r;
      if (m < n_nodes) dstp[m] = acc[r];
    }
  }
}

// ---------------------------------------------------------------------------
// Kernel 2: per-edge gather + sigmoid + weight. Bandwidth/gather bound;
// proj arrays (400 KB) live in L2 (192 MB), so random gathers are cheap.
// ---------------------------------------------------------------------------
__global__ void __launch_bounds__(256)
edge_score(const long long* __restrict__ src_idx,
           const long long* __restrict__ dst_idx,
           const float* __restrict__ edge_w,
           const float* __restrict__ proj_src,
           const float* __restrict__ proj_dst,
           const float* __restrict__ bias,
           float* __restrict__ out,
           int n_edges) {
  const int e = blockIdx.x * blockDim.x + threadIdx.x;
  if (e >= n_edges) return;
  const long long s = src_idx[e];
  const long long d = dst_idx[e];
  const float logit = proj_src[s] + proj_dst[d] + bias[0];
  const float sig = 1.0f / (1.0f + __expf(-logit));
  out[e] = edge_w[e] * sig;
}

extern "C" void kernel_launch(void* const* d_in, const int* in_sizes, int n_in,
                              void* d_out, int out_size, void* d_ws, size_t ws_size,
                              hipStream_t stream) {
  // Inputs per setup_inputs() order:
  //  0: x           [N_NODES * 128]  float32
  //  1: edge_index  [2 * N_EDGES]    int64
  //  2: edge_weight [N_EDGES]        float32
  //  3: W           [256]            float32
  //  4: b           [1]              float32
  const float*     x  = (const float*)d_in[0];
  const long long* ei = (const long long*)d_in[1];
  const float*     ew = (const float*)d_in[2];
  const float*     W  = (const float*)d_in[3];
  const float*     b  = (const float*)d_in[4];
  float* out = (float*)d_out;

  const int n_nodes = in_sizes[0] / IN_DIM;
  const int n_edges = in_sizes[2];

  const long long* src_idx = ei;             // edge_index[0]
  const long long* dst_idx = ei + n_edges;   // edge_index[1]

  float* proj_src = (float*)d_ws;            // [n_nodes]
  float* proj_dst = proj_src + n_nodes;      // [n_nodes]

  const int n_tiles = (n_nodes + 15) / 16;
  const int waves_per_block = 8;             // 256 threads = 8 wave32
  const int grid1 = (n_tiles + waves_per_block - 1) / waves_per_block;
  node_proj_wmma<<<grid1, 256, 0, stream>>>(x, W, proj_src, proj_dst,
                                            n_nodes, n_tiles);

  const int grid2 = (n_edges + 255) / 256;
  edge_score<<<grid2, 256, 0, stream>>>(src_idx, dst_idx, ew,
                                        proj_src, proj_dst, b, out, n_edges);
}